// QAoutputBlock_40785009443711
// MI455X (gfx1250) — compile-verified
//
#include <hip/hip_runtime.h>

// ---------------------------------------------------------------------------
// QA output block: banded outer-product argmax (start/end span extraction).
// B=32, S=2048, band width = ans_limit+1 = 31.
//
// MI455X reasoning: total input is 512 KB -> ~22 ns at 23.3 TB/s HBM; this is
// launch/latency bound. One workgroup per batch, both rows staged in LDS.
// Each 16-row band tile is computed TWICE on the WMMA pipe:
//   dF = x1_tile (M=i) x x2_seg (N=j)   -> column (end-index) maxima
//   dT = x2_seg (M=j) x x1_tile (N=i)   -> row (start-index) maxima
// so BOTH reductions are per-lane selects over the 8 accumulator VGPRs plus
// one xor-16 lane combine (no 16-lane butterfly). Reductions are
// select-then-tree (log-depth) so dual-issue VALU can pack them. Operands are
// built with a lane-factor multiply so ds_loads stay unpredicated. Full f32
// WMMA keeps argmax bit-identical to the f32 reference. Band masks are
// tile-invariant -> hoisted into SGPR bitmasks.
// ---------------------------------------------------------------------------

typedef __attribute__((ext_vector_type(2))) float v2f;
typedef __attribute__((ext_vector_type(8))) float v8f;

#define SDIM       2048
#define NTHREADS   256
#define NWAVES     8            // 256 threads / wave32
#define NTILES     (SDIM / 16)  // 128 row tiles
#define TPW        (NTILES / NWAVES)
#define PAD        48           // x2 tail padding: last tile reads j up to base+47

// Order-preserving float -> uint encoding (monotone: f1 < f2  <=>  e1 < e2)
__device__ __forceinline__ unsigned enc_f32(float f) {
    unsigned u = __float_as_uint(f);
    return (u & 0x80000000u) ? ~u : (u | 0x80000000u);
}
__device__ __forceinline__ float dec_f32(unsigned e) {
    unsigned u = (e & 0x80000000u) ? (e & 0x7FFFFFFFu) : ~e;
    return __uint_as_float(u);
}

// lexicographic (value desc, index asc) combine: associative + commutative
__device__ __forceinline__ void better(float& v, int& i, float ov, int oi) {
    if (ov > v || (ov == v && oi < i)) { v = ov; i = oi; }
}

// log-depth in-place max fold of N elements
template <int N>
__device__ __forceinline__ float treemax(float* t) {
#pragma unroll
    for (int st = N; st > 1; st = (st + 1) / 2) {
#pragma unroll
        for (int k = 0; k < st / 2; ++k)
            t[k] = fmaxf(t[k], t[st - 1 - k]);
    }
    return t[0];
}

__global__ __launch_bounds__(NTHREADS)
void qa_band_argmax_kernel(const float* __restrict__ x1,
                           const float* __restrict__ x2,
                           const int* __restrict__ ans_limit_p,
                           float* __restrict__ out, int B) {
    __shared__ float    x1s[SDIM];
    __shared__ float    x2s[SDIM + PAD];
    __shared__ float    rowmax[SDIM];
    __shared__ unsigned colmax[SDIM];          // enc_f32 encoded
    __shared__ float    rvs[NWAVES], cvs[NWAVES];
    __shared__ int      ris[NWAVES], cis[NWAVES];

    const int b    = blockIdx.x;
    const int tid  = threadIdx.x;
    const int lane = tid & 31;
    const int wave = tid >> 5;
    const int n    = lane & 15;    // N position within 16x16 tile
    const int hi   = lane >> 4;    // 0: rows 0..7 half, 1: rows 8..15 half
    const int limit = ans_limit_p[0];
    const float NEG = -__builtin_inff();
    const float kz  = hi ? 0.0f : 1.0f;   // lane factor: zero K=2 slots by multiply

    const float* __restrict__ x1row = x1 + (size_t)b * SDIM;
    const float* __restrict__ x2row = x2 + (size_t)b * SDIM;

    // gfx1250 speculative prefetch of this block's two rows
    __builtin_prefetch(x1row + tid * 8, 0, 0);
    __builtin_prefetch(x2row + tid * 8, 0, 0);

    // ---- stage rows into LDS (float4, 2 vectors per thread per row) ----
    {
        const float4* x1v = (const float4*)x1row;
        const float4* x2v = (const float4*)x2row;
        float4* x1sv = (float4*)x1s;
        float4* x2sv = (float4*)x2s;
#pragma unroll
        for (int k = 0; k < SDIM / 4 / NTHREADS; ++k) {
            x1sv[tid + k * NTHREADS] = x1v[tid + k * NTHREADS];
            x2sv[tid + k * NTHREADS] = x2v[tid + k * NTHREADS];
        }
        if (tid < PAD) x2s[SDIM + tid] = 0.0f;     // zero pad: products clamp at 0 anyway
#pragma unroll
        for (int k = 0; k < SDIM / NTHREADS; ++k)
            colmax[tid + k * NTHREADS] = 0x80000000u;  // enc(+0.0f): band max clamps at 0
    }

    // ---- tile-invariant band masks (hoisted -> SGPR wave masks) ----
    // dF element (m = r+8*hi, n): i = base+m, j = base+16c+n, delta = 16c+n-m
    // dT element (m = r+8*hi, n): j = base+16c+m, i = base+n, delta = 16c+m-n
    bool maskF[3][8], maskT[3][8];
#pragma unroll
    for (int c = 0; c < 3; ++c)
#pragma unroll
        for (int r = 0; r < 8; ++r) {
            const int m  = r + hi * 8;
            const int dF = c * 16 + n - m;
            const int dT = c * 16 + m - n;
            maskF[c][r] = (dF >= 0) & (dF <= limit);
            maskT[c][r] = (dT >= 0) & (dT <= limit);
        }

    __syncthreads();

    // ---- per-wave loop over 16-row tiles ----
    for (int tl = 0; tl < TPW; ++tl) {
        const int base = (wave * TPW + tl) * 16;

        // Unconditional LDS loads; lane-factor multiply zeroes the K=2 slots of
        // lanes 16-31 (finite*0 = +-0, contributes +-0 to the WMMA dot: exact).
        v2f av;
        av.x = x1s[base + n] * kz;
        av.y = 0.0f;

        v8f dF[3], dT[3];
#pragma unroll
        for (int c = 0; c < 3; ++c) {
            v2f bv;
            bv.x = x2s[base + c * 16 + n] * kz;
            bv.y = 0.0f;
            v8f cz = {};
            // dF[m,n] = x1[base+m] * x2[base+16c+n]
            dF[c] = __builtin_amdgcn_wmma_f32_16x16x4_f32(
                false, av, false, bv, (short)0, cz, false, false);
            v8f cz2 = {};
            // dT[m,n] = x2[base+16c+m] * x1[base+n]   (transposed orientation)
            dT[c] = __builtin_amdgcn_wmma_f32_16x16x4_f32(
                false, bv, false, av, (short)0, cz2, false, false);
        }

        // ---- row maxima (start index i = base+n): 24 independent masked
        //      selects, log-depth tree, one xor-16 combine, wave-owned store
        {
            float t[24];
#pragma unroll
            for (int c = 0; c < 3; ++c)
#pragma unroll
                for (int r = 0; r < 8; ++r)
                    t[c * 8 + r] = maskT[c][r] ? dT[c][r] : NEG;
            float rv = treemax<24>(t);
            rv = fmaxf(rv, __shfl_xor(rv, 16, 32));
            if (hi == 0)
                rowmax[base + n] = rv;
        }

        // ---- column maxima (end index j = base+16c+n): select + tree over
        //      8 VGPRs, xor-16 combine, ds_max_u32 into shared table
#pragma unroll
        for (int c = 0; c < 3; ++c) {
            float t[8];
#pragma unroll
            for (int r = 0; r < 8; ++r)
                t[r] = maskF[c][r] ? dF[c][r] : NEG;
            float v = treemax<8>(t);
            v = fmaxf(v, __shfl_xor(v, 16, 32));
            const int j = base + c * 16 + n;
            if (hi == 0 && j < SDIM)
                atomicMax(&colmax[j], enc_f32(v));   // enc(-inf) < enc(0): no-op if empty
        }
    }
    __syncthreads();

    // ---- block argmax over rowmax (clamped at 0) and colmax, first-index ties ----
    float bv1 = NEG, bv2 = NEG;
    int bi1 = 0, bi2 = 0;
#pragma unroll
    for (int k = 0; k < SDIM / NTHREADS; ++k) {
        const int idx = tid + k * NTHREADS;
        const float r = fmaxf(rowmax[idx], 0.0f);  // zeros outside band join the max
        better(bv1, bi1, r, idx);
        const float cv = dec_f32(colmax[idx]);     // already >= 0 by init
        better(bv2, bi2, cv, idx);
    }
    // in-wave shuffle reduction (lex order is associative/commutative)
#pragma unroll
    for (int s = 16; s > 0; s >>= 1) {
        float ov = __shfl_xor(bv1, s, 32); int oi = __shfl_xor(bi1, s, 32);
        better(bv1, bi1, ov, oi);
        ov = __shfl_xor(bv2, s, 32); oi = __shfl_xor(bi2, s, 32);
        better(bv2, bi2, ov, oi);
    }
    if (lane == 0) { rvs[wave] = bv1; ris[wave] = bi1; cvs[wave] = bv2; cis[wave] = bi2; }
    __syncthreads();
    if (tid == 0) {
        float v1 = rvs[0]; int i1 = ris[0];
        float v2 = cvs[0]; int i2 = cis[0];
#pragma unroll
        for (int w = 1; w < NWAVES; ++w) {
            better(v1, i1, rvs[w], ris[w]);
            better(v2, i2, cvs[w], cis[w]);
        }
        out[b]     = (float)i1;   // out1: best start index
        out[B + b] = (float)i2;   // out2: best end index
    }
}

extern "C" void kernel_launch(void* const* d_in, const int* in_sizes, int n_in,
                              void* d_out, int out_size, void* d_ws, size_t ws_size,
                              hipStream_t stream) {
    const float* x1  = (const float*)d_in[0];
    const float* x2  = (const float*)d_in[1];
    const int*   lim = (const int*)d_in[2];
    float* out = (float*)d_out;
    const int B = in_sizes[0] / SDIM;   // 32
    qa_band_argmax_kernel<<<dim3(B), dim3(NTHREADS), 0, stream>>>(x1, x2, lim, out, B);
}